// GINClassifier_26491358282142
// MI455X (gfx1250) — compile-verified
//
#include <hip/hip_runtime.h>
#include <hip/hip_bf16.h>

// ---------------------------------------------------------------------------
// GIN classifier for MI455X (gfx1250, wave32).
//
// Roofline: GEMMs total ~5.7 GFLOP (negligible on matrix cores; fp32 WMMA
// V_WMMA_F32_16X16X4_F32 keeps reference precision). Dominant cost is edge
// aggregation. Instead of E*D fp32 scatter atomics (~820M RMWs), we build a
// by-destination CSR once (~6.4M int atomics + 100K scan) and aggregate by
// pure gather: node features are L2-resident (26-51 MB << 192 MB L2), each
// output row is written exactly once, and the GIN self-term (x + agg) is
// fused into the gather.
// ---------------------------------------------------------------------------

typedef __attribute__((ext_vector_type(2))) float v2f;
typedef __attribute__((ext_vector_type(8))) float v8f;

#define HID 64
#define BN_EPS 1e-5f

__device__ __forceinline__ v8f wmma_f32(v2f a, v2f b, v8f c) {
    // D = A(16x4 f32) * B(4x16 f32) + C(16x16 f32)
    return __builtin_amdgcn_wmma_f32_16x16x4_f32(
        /*neg_a=*/false, a, /*neg_b=*/false, b,
        /*c_mod=*/(short)0, c, /*reuse_a=*/false, /*reuse_b=*/false);
}

// Wave computes a 16x64 output tile: rows [m0, m0+16) of shA (lda-strided,
// local rows), all 64 columns of row-major KxN matrix B.
// A layout (ISA 7.12.2, 32-bit A 16x4): lane = M; VGPR pair = K pair; half-
// wave selects K offset 0/2.  B layout: VGPRv = K row v (lanes 0-15) and
// K row v+2 (lanes 16-31), N = lane&15.
__device__ __forceinline__ void wave_gemm_64(const float* shA, int lda, int m0,
                                             const float* B, int ldb, int K,
                                             int lane, v8f c[4]) {
    const int lm = lane & 15;
    const int koff = (lane >> 4) * 2;
    for (int k0 = 0; k0 < K; k0 += 4) {
        v2f a;
        a.x = shA[(m0 + lm) * lda + k0 + koff];
        a.y = shA[(m0 + lm) * lda + k0 + koff + 1];
        const float* B0 = B + (size_t)(k0 + koff) * ldb;
        const float* B1 = B + (size_t)(k0 + koff + 1) * ldb;
#pragma unroll
        for (int t = 0; t < 4; ++t) {
            v2f b;
            b.x = B0[t * 16 + lm];
            b.y = B1[t * 16 + lm];
            c[t] = wmma_f32(a, b, c[t]);
        }
    }
}

// -------------------------- small helpers ----------------------------------

__global__ void zero_f32_kernel(float* __restrict__ p, int n) {
    int t = blockIdx.x * blockDim.x + threadIdx.x;
    if (t < n) p[t] = 0.0f;
}

__global__ void zero_i32_kernel(int* __restrict__ p, int n) {
    int t = blockIdx.x * blockDim.x + threadIdx.x;
    if (t < n) p[t] = 0;
}

// -------------------------- CSR construction -------------------------------

__global__ void count_kernel(const int* __restrict__ dst, int* __restrict__ deg, int E) {
    int t = blockIdx.x * blockDim.x + threadIdx.x;
    if (t < E) atomicAdd(&deg[dst[t]], 1);
}

// Single-workgroup chunked exclusive scan over deg -> offs (and cursor copy).
__global__ void __launch_bounds__(256)
scan_kernel(const int* __restrict__ deg, int* __restrict__ offs,
            int* __restrict__ cursor, int N) {
    __shared__ int sh[256];
    __shared__ int carry;
    const int tid = threadIdx.x;
    if (tid == 0) carry = 0;
    __syncthreads();
    for (int base = 0; base < N; base += 256) {
        const int i = base + tid;
        const int v = (i < N) ? deg[i] : 0;
        sh[tid] = v;
        __syncthreads();
        for (int off = 1; off < 256; off <<= 1) {
            int add = (tid >= off) ? sh[tid - off] : 0;
            __syncthreads();
            sh[tid] += add;
            __syncthreads();
        }
        const int incl = sh[tid];
        const int c = carry;
        if (i < N) {
            offs[i] = c + incl - v;
            cursor[i] = c + incl - v;
        }
        __syncthreads();
        if (tid == 255) carry = c + incl;
        __syncthreads();
    }
    if (tid == 0) offs[N] = carry;
}

// csr_src[cursor[dst[e]]++] = src[e]
__global__ void fill_kernel(const int* __restrict__ src, const int* __restrict__ dst,
                            int* __restrict__ cursor, int* __restrict__ csr_src, int E) {
    int t = blockIdx.x * blockDim.x + threadIdx.x;
    if (t >= E) return;
    int d = dst[t];
    int p = atomicAdd(&cursor[d], 1);
    csr_src[p] = src[t];
}

// hsum[n] = x[n] + sum_{j in in-edges(n)} x[src_j]   (pure gather, no atomics)
__global__ void gather_sum_kernel(const float* __restrict__ x,
                                  const int* __restrict__ offs,
                                  const int* __restrict__ csr_src,
                                  float* __restrict__ hsum, int N, int D) {
    int tid = blockIdx.x * blockDim.x + threadIdx.x;
    int vec = D >> 2;
    if (tid >= N * vec) return;
    int node = tid / vec;
    int c = (tid - node * vec) << 2;
    float4 acc = *reinterpret_cast<const float4*>(x + (size_t)node * D + c);  // self term
    int j1 = offs[node + 1];
    for (int j = offs[node]; j < j1; ++j) {
        int s = csr_src[j];
        const float4 v = *reinterpret_cast<const float4*>(x + (size_t)s * D + c);
        acc.x += v.x; acc.y += v.y; acc.z += v.z; acc.w += v.w;
    }
    *reinterpret_cast<float4*>(hsum + (size_t)node * D + c) = acc;
}

// -------------------------- elementwise ------------------------------------

// x' = relu(gamma*(h-mu)*rsqrt(var+eps)+beta), biased var (matches jnp.var)
__global__ void bn_apply_kernel(const float* __restrict__ h,
                                const float* __restrict__ stats,
                                const float* __restrict__ gamma,
                                const float* __restrict__ beta,
                                float* __restrict__ out, int N) {
    int tid = blockIdx.x * blockDim.x + threadIdx.x;
    if (tid >= N * HID) return;
    int c = tid & (HID - 1);
    float invN = 1.0f / (float)N;
    float mu = stats[c] * invN;
    float var = stats[HID + c] * invN - mu * mu;
    float v = gamma[c] * (h[tid] - mu) * rsqrtf(var + BN_EPS) + beta[c];
    out[tid] = v > 0.0f ? v : 0.0f;
}

// hg[batch[n]] += x[n]  (only N*HID atomics total — cheap)
__global__ void pool_kernel(const float* __restrict__ x,
                            const int* __restrict__ batch,
                            float* __restrict__ hg, int N) {
    int tid = blockIdx.x * blockDim.x + threadIdx.x;
    if (tid >= N * (HID / 4)) return;
    int node = tid >> 4;
    int c = (tid & 15) << 2;
    int g = batch[node];
    const float4 v = *reinterpret_cast<const float4*>(x + (size_t)node * HID + c);
    float* p = hg + (size_t)g * HID + c;
    atomicAdd(p + 0, v.x);
    atomicAdd(p + 1, v.y);
    atomicAdd(p + 2, v.z);
    atomicAdd(p + 3, v.w);
}

// -------------------------- fused GIN MLP (WMMA) ---------------------------
// hout = relu(hpre @ w1 + b1) @ w2 + b2; accumulates per-column sum/sumsq.
// 4 waves/WG, 64 rows/WG.
__global__ void __launch_bounds__(128)
gin_mlp_kernel(const float* __restrict__ hpre,
               const float* __restrict__ w1, const float* __restrict__ b1,
               const float* __restrict__ w2, const float* __restrict__ b2,
               float* __restrict__ hout, float* __restrict__ stats,
               int N, int Din) {
    __shared__ float sh_h[64 * 128];   // input tile, Din <= 128
    __shared__ float sh_t[64 * HID];   // relu(h@w1+b1) tile
    __shared__ float sh_sum[HID];
    __shared__ float sh_sq[HID];

    const int tid = threadIdx.x;
    const int row0 = blockIdx.x * 64;

    if (tid < HID) { sh_sum[tid] = 0.0f; sh_sq[tid] = 0.0f; }

    // Stage input tile (rows >= N zero-padded so WMMA runs with full EXEC).
    const int elems = 64 * Din;
    for (int i = tid; i < elems; i += 128) {
        int r = i / Din, k = i - r * Din;
        int gr = row0 + r;
        sh_h[r * Din + k] = (gr < N) ? hpre[(size_t)gr * Din + k] : 0.0f;
    }
    __syncthreads();

    const int wave = tid >> 5;
    const int lane = tid & 31;
    const int m0 = wave * 16;
    const int lm = lane & 15;
    const int hi = lane >> 4;

    // GEMM1: t = relu(h @ w1 + b1)
    v8f c[4] = {};
    wave_gemm_64(sh_h, Din, m0, w1, HID, Din, lane, c);
#pragma unroll
    for (int t = 0; t < 4; ++t) {
        int n = t * 16 + lm;
        float bias = b1[n];
#pragma unroll
        for (int i = 0; i < 8; ++i) {
            float v = c[t][i] + bias;                  // C/D: row m0+hi*8+i
            v = v > 0.0f ? v : 0.0f;
            sh_t[(m0 + hi * 8 + i) * HID + n] = v;
        }
    }
    __syncthreads();

    // GEMM2: h_out = t @ w2 + b2
    v8f d[4] = {};
    wave_gemm_64(sh_t, HID, m0, w2, HID, HID, lane, d);
#pragma unroll
    for (int t = 0; t < 4; ++t) {
        int n = t * 16 + lm;
        float bias = b2[n];
        float s = 0.0f, q = 0.0f;
#pragma unroll
        for (int i = 0; i < 8; ++i) {
            int gm = row0 + m0 + hi * 8 + i;
            float v = d[t][i] + bias;
            if (gm < N) {
                hout[(size_t)gm * HID + n] = v;
                s += v;
                q += v * v;
            }
        }
        atomicAdd(&sh_sum[n], s);
        atomicAdd(&sh_sq[n], q);
    }
    __syncthreads();
    if (tid < HID) {
        atomicAdd(&stats[tid], sh_sum[tid]);
        atomicAdd(&stats[HID + tid], sh_sq[tid]);
    }
}

// -------------------------- head MLP (WMMA) --------------------------------
// out = relu(hg @ w1 + b1) @ w2 + b2, w2 is 64x10 (zero-padded to 16 cols).
__global__ void __launch_bounds__(128)
head_kernel(const float* __restrict__ hg, const float* __restrict__ w1,
            const float* __restrict__ b1, const float* __restrict__ w2,
            const float* __restrict__ b2, float* __restrict__ out) {
    __shared__ float sh_h[64 * HID];
    __shared__ float sh_t[64 * HID];
    __shared__ float sh_w2[HID * 16];

    const int tid = threadIdx.x;
    const int row0 = blockIdx.x * 64;   // G=512 divisible by 64

    for (int i = tid; i < 64 * HID; i += 128) sh_h[i] = hg[(size_t)row0 * HID + i];
    for (int i = tid; i < HID * 16; i += 128) {
        int k = i >> 4, n = i & 15;
        sh_w2[i] = (n < 10) ? w2[k * 10 + n] : 0.0f;
    }
    __syncthreads();

    const int wave = tid >> 5;
    const int lane = tid & 31;
    const int m0 = wave * 16;
    const int lm = lane & 15;
    const int hi = lane >> 4;
    const int koff = hi * 2;

    v8f c[4] = {};
    wave_gemm_64(sh_h, HID, m0, w1, HID, HID, lane, c);
#pragma unroll
    for (int t = 0; t < 4; ++t) {
        int n = t * 16 + lm;
        float bias = b1[n];
#pragma unroll
        for (int i = 0; i < 8; ++i) {
            float v = c[t][i] + bias;
            v = v > 0.0f ? v : 0.0f;
            sh_t[(m0 + hi * 8 + i) * HID + n] = v;
        }
    }
    __syncthreads();

    v8f d = {};
    for (int k0 = 0; k0 < HID; k0 += 4) {
        v2f a, b;
        a.x = sh_t[(m0 + lm) * HID + k0 + koff];
        a.y = sh_t[(m0 + lm) * HID + k0 + koff + 1];
        b.x = sh_w2[(k0 + koff) * 16 + lm];
        b.y = sh_w2[(k0 + koff + 1) * 16 + lm];
        d = wmma_f32(a, b, d);
    }
    if (lm < 10) {
        float bias = b2[lm];
#pragma unroll
        for (int i = 0; i < 8; ++i)
            out[(size_t)(row0 + m0 + hi * 8 + i) * 10 + lm] = d[i] + bias;
    }
}

// ---------------------------------------------------------------------------

extern "C" void kernel_launch(void* const* d_in, const int* in_sizes, int n_in,
                              void* d_out, int out_size, void* d_ws, size_t ws_size,
                              hipStream_t stream) {
    (void)in_sizes; (void)n_in; (void)out_size; (void)ws_size;
    constexpr int N = 100000, E = 3200000, F = 128, G = 512;

    const float* x     = (const float*)d_in[0];
    const int*   ei    = (const int*)d_in[1];
    const int*   batch = (const int*)d_in[2];
    const int* src = ei;
    const int* dst = ei + E;

    const float* w1[3] = {(const float*)d_in[3],  (const float*)d_in[9],  (const float*)d_in[15]};
    const float* b1[3] = {(const float*)d_in[4],  (const float*)d_in[10], (const float*)d_in[16]};
    const float* w2[3] = {(const float*)d_in[5],  (const float*)d_in[11], (const float*)d_in[17]};
    const float* b2[3] = {(const float*)d_in[6],  (const float*)d_in[12], (const float*)d_in[18]};
    const float* gam[3]= {(const float*)d_in[7],  (const float*)d_in[13], (const float*)d_in[19]};
    const float* bet[3]= {(const float*)d_in[8],  (const float*)d_in[14], (const float*)d_in[20]};
    const float* hw1 = (const float*)d_in[21];
    const float* hb1 = (const float*)d_in[22];
    const float* hw2 = (const float*)d_in[23];
    const float* hb2 = (const float*)d_in[24];
    float* out = (float*)d_out;

    // Workspace layout:
    //   ints:   deg[N] | offs[N+1] | cursor[N] | csr_src[E]
    //   floats: hsum[N*128] | hbuf[N*64] | xbuf[N*64] | stats[128] | hg[G*64]
    int* deg    = (int*)d_ws;
    int* offs   = deg + N;
    int* cursor = offs + (N + 1);
    int* csr    = cursor + N;
    uintptr_t pa = ((uintptr_t)(csr + E) + 15u) & ~(uintptr_t)15u;  // align for float4
    float* hsum  = (float*)pa;
    float* hbuf  = hsum + (size_t)N * F;
    float* xbuf  = hbuf + (size_t)N * HID;
    float* stats = xbuf + (size_t)N * HID;
    float* hg    = stats + 2 * HID;

    // ---- COO -> CSR (by destination), built once, reused by all 3 layers
    zero_i32_kernel<<<(N + 255) / 256, 256, 0, stream>>>(deg, N);
    count_kernel<<<(E + 255) / 256, 256, 0, stream>>>(dst, deg, E);
    scan_kernel<<<1, 256, 0, stream>>>(deg, offs, cursor, N);
    fill_kernel<<<(E + 255) / 256, 256, 0, stream>>>(src, dst, cursor, csr, E);

    const float* xin = x;
    for (int l = 0; l < 3; ++l) {
        int Din = (l == 0) ? F : HID;
        int ngt = N * (Din / 4);
        gather_sum_kernel<<<(ngt + 255) / 256, 256, 0, stream>>>(xin, offs, csr, hsum, N, Din);
        zero_f32_kernel<<<1, 256, 0, stream>>>(stats, 2 * HID);
        gin_mlp_kernel<<<(N + 63) / 64, 128, 0, stream>>>(hsum, w1[l], b1[l], w2[l], b2[l],
                                                          hbuf, stats, N, Din);
        bn_apply_kernel<<<(N * HID + 255) / 256, 256, 0, stream>>>(hbuf, stats, gam[l], bet[l],
                                                                   xbuf, N);
        xin = xbuf;
    }
    zero_f32_kernel<<<(G * HID + 255) / 256, 256, 0, stream>>>(hg, G * HID);
    pool_kernel<<<(N * (HID / 4) + 255) / 256, 256, 0, stream>>>(xbuf, batch, hg, N);
    head_kernel<<<G / 64, 128, 0, stream>>>(hg, hw1, hb1, hw2, hb2, out);
}